// FilterLayer_89532888252825
// MI455X (gfx1250) — compile-verified
//
#include <hip/hip_runtime.h>
#include <hip/hip_bf16.h>
#include <stdint.h>

typedef float v2f __attribute__((ext_vector_type(2)));
typedef float v4f __attribute__((ext_vector_type(4)));
typedef float v8f __attribute__((ext_vector_type(8)));

#define SEQ   512
#define NH    256
#define NB    256
#define KR    264          // stored freq rows: 257 live + 7 zeroed (33 x 8-k steps)
#define CHUNK 32           // batches per chunk: Zr/Zi (~17.3 MB) stays L2-resident

#define FWD_ELEMS (16 * 64 * 32 * 4)   // 131072 = 256x512, fragment-swizzled
#define INV_ELEMS (32 * 33 * 32 * 4)   // 135168 = 512x264, fragment-swizzled

#define ORTHO_SCALE 0.044194173824159216f   // 1/sqrt(512)

#define WMMA_F32(A, Bf, C) \
    __builtin_amdgcn_wmma_f32_16x16x4_f32(false, (A), false, (Bf), (short)0, (C), false, false)

// ---------------------------------------------------------------------------
// Build DFT bases directly in WMMA-fragment order:
//   layout [tile][i = kstep pair][lane][4]:
//     floats 0,1 -> lane's A pair for k-step 8i,   floats 2,3 -> step 8i+4
//     lane<16 supplies K=+0,+1 ; lane>=16 supplies K=+2,+3 (ISA 16x4 A layout)
// Forward (M = freq k, contraction = s):  Fc =  cos/sqrtS, Fs = -sin/sqrtS
// Inverse (M = s, contraction = k):       Gc = c_k*cos/sqrtS, Gs = -d_k*sin/sqrtS
//   c = {1, 2..2, 1, 0 x7},  d = {0, 2..2, 0, 0 x7}  (irfft drops DC/Nyquist imag)
// ---------------------------------------------------------------------------
__global__ __launch_bounds__(256) void build_dft(float* __restrict__ fcS,
                                                 float* __restrict__ fsS,
                                                 float* __restrict__ gcS,
                                                 float* __restrict__ gsS) {
    const int i = blockIdx.x * blockDim.x + threadIdx.x;
    const float w0 = 6.28318530717958647692f / (float)SEQ;

    if (i < FWD_ELEMS) {
        const int q    = i & 3;
        const int ln   = (i >> 2) & 31;
        const int rest = i >> 7;
        const int it   = rest & 63;       // k-step pair index
        const int t    = rest >> 6;       // k tile 0..15
        const int k    = t * 16 + (ln & 15);
        const int s    = 8 * it + ((q >> 1) << 2) + 2 * (ln >> 4) + (q & 1);
        const int m    = (k * s) & (SEQ - 1);
        float sn, cs;
        sincosf(w0 * (float)m, &sn, &cs);
        fcS[i] = cs * ORTHO_SCALE;
        fsS[i] = -sn * ORTHO_SCALE;
    }
    if (i < INV_ELEMS) {
        const int q    = i & 3;
        const int ln   = (i >> 2) & 31;
        const int rest = i >> 7;          // 0..1055
        const int it   = rest % 33;       // k-step pair index 0..32
        const int tS   = rest / 33;       // s tile 0..31
        const int s    = tS * 16 + (ln & 15);
        const int k    = 8 * it + ((q >> 1) << 2) + 2 * (ln >> 4) + (q & 1);
        const int m    = (k * s) & (SEQ - 1);
        float sn, cs;
        sincosf(w0 * (float)m, &sn, &cs);
        float ck = 0.0f, dk = 0.0f;
        if (k == 0)        { ck = 1.0f; dk = 0.0f; }
        else if (k < 256)  { ck = 2.0f; dk = 2.0f; }
        else if (k == 256) { ck = 1.0f; dk = 0.0f; }
        gcS[i] = ck * cs * ORTHO_SCALE;
        gsS[i] = -dk * sn * ORTHO_SCALE;
    }
}

// ---------------------------------------------------------------------------
// Stage A: Xr = Fc @ x_b, Xi = Fs @ x_b per batch, fused Z = X*W epilogue.
// x tile (512x16, 32 KB) staged via global_load_async_to_lds_b128 (ASYNCcnt);
// 8 waves x 2 k-tiles; software-pipelined: iteration i preloads i+1's
// fragments, then runs 8 WMMAs.
// ---------------------------------------------------------------------------
__global__ __launch_bounds__(256) void stage_a(const float* __restrict__ x,
                                               const float* __restrict__ w,
                                               const float* __restrict__ fcS,
                                               const float* __restrict__ fsS,
                                               float* __restrict__ zr,
                                               float* __restrict__ zi,
                                               int b0) {
    __shared__ float xs[SEQ * 16];       // [s][16h], 32 KB
    __shared__ float red[16 * 16];       // Nyquist reduction scratch

    const int hT   = blockIdx.x;         // 0..15 h tile
    const int bb   = blockIdx.y;         // 0..CHUNK-1
    const int b    = b0 + bb;
    const int tid  = threadIdx.x;
    const int wv   = tid >> 5;           // wave 0..7 -> k tiles wv, wv+8
    const int lane = tid & 31;
    const int lm   = lane & 15;
    const int hi   = lane >> 4;

    const int h0 = hT * 16;
    const float* __restrict__ xb = x + (size_t)b * SEQ * NH + h0;

    // ---- async stage x tile into LDS: 2048 x 16B transfers, coalesced ----
#pragma unroll
    for (int j = 0; j < 8; ++j) {
        const int u   = tid + 256 * j;   // 16-byte unit index
        const int s   = u >> 2;
        const int seg = u & 3;
        const uint32_t la = (uint32_t)(uintptr_t)(&xs[s * 16 + seg * 4]);
        const uint64_t ga = (uint64_t)(uintptr_t)(xb + (size_t)s * NH + seg * 4);
        asm volatile("global_load_async_to_lds_b128 %0, %1, off"
                     :: "v"(la), "v"(ga) : "memory");
    }
    asm volatile("s_wait_asynccnt 0x0" ::: "memory");
    __syncthreads();

    // ---- pipelined WMMA GEMM: 2 k-tiles x (R,I) x 2 steps = 8 WMMA/iter ----
    const int t0 = wv, t1 = wv + 8;
    const float* __restrict__ pc0 = fcS + (size_t)t0 * 8192 + (size_t)lane * 4;
    const float* __restrict__ ps0 = fsS + (size_t)t0 * 8192 + (size_t)lane * 4;
    const float* __restrict__ pc1 = fcS + (size_t)t1 * 8192 + (size_t)lane * 4;
    const float* __restrict__ ps1 = fsS + (size_t)t1 * 8192 + (size_t)lane * 4;
    const int bbase = 2 * hi * 16 + lm;  // xs index base for a k-step

    v8f aR0 = {0.f,0.f,0.f,0.f,0.f,0.f,0.f,0.f};
    v8f aI0 = {0.f,0.f,0.f,0.f,0.f,0.f,0.f,0.f};
    v8f aR1 = {0.f,0.f,0.f,0.f,0.f,0.f,0.f,0.f};
    v8f aI1 = {0.f,0.f,0.f,0.f,0.f,0.f,0.f,0.f};

    v4f C0 = *(const v4f*)(pc0), S0 = *(const v4f*)(ps0);
    v4f C1 = *(const v4f*)(pc1), S1 = *(const v4f*)(ps1);
    v2f B0, B1;
    B0.x = xs[bbase];      B0.y = xs[bbase + 16];
    B1.x = xs[bbase + 64]; B1.y = xs[bbase + 80];

    for (int i = 0; i < 64; ++i) {
        const int ip  = (i < 63) ? i + 1 : 63;       // uniform clamp
        const int off = ip * 128;
        v4f nC0 = *(const v4f*)(pc0 + off), nS0 = *(const v4f*)(ps0 + off);
        v4f nC1 = *(const v4f*)(pc1 + off), nS1 = *(const v4f*)(ps1 + off);
        const int xo = off + bbase;                   // kk = 8*ip -> xs row *16
        v2f nB0, nB1;
        nB0.x = xs[xo];      nB0.y = xs[xo + 16];
        nB1.x = xs[xo + 64]; nB1.y = xs[xo + 80];

        v2f a;
        a = __builtin_shufflevector(C0, C0, 0, 1); aR0 = WMMA_F32(a, B0, aR0);
        a = __builtin_shufflevector(S0, S0, 0, 1); aI0 = WMMA_F32(a, B0, aI0);
        a = __builtin_shufflevector(C1, C1, 0, 1); aR1 = WMMA_F32(a, B0, aR1);
        a = __builtin_shufflevector(S1, S1, 0, 1); aI1 = WMMA_F32(a, B0, aI1);
        a = __builtin_shufflevector(C0, C0, 2, 3); aR0 = WMMA_F32(a, B1, aR0);
        a = __builtin_shufflevector(S0, S0, 2, 3); aI0 = WMMA_F32(a, B1, aI0);
        a = __builtin_shufflevector(C1, C1, 2, 3); aR1 = WMMA_F32(a, B1, aR1);
        a = __builtin_shufflevector(S1, S1, 2, 3); aI1 = WMMA_F32(a, B1, aI1);

        C0 = nC0; S0 = nS0; C1 = nC1; S1 = nS1; B0 = nB0; B1 = nB1;
    }

    // ---- epilogue: Z = X * W (complex); k rows here are all < 256 ----
    const int hcol = h0 + lm;
    float* __restrict__ zrB = zr + (size_t)bb * KR * NH;
    float* __restrict__ ziB = zi + (size_t)bb * KR * NH;
#pragma unroll
    for (int r = 0; r < 8; ++r) {
        const int m = r + 8 * hi;        // C/D layout: lanes>=16 hold M=8..15
        {
            const int krow = t0 * 16 + m;
            const size_t wix = ((size_t)krow * NH + hcol) * 2;
            const float wr = w[wix], wi = w[wix + 1];
            zrB[(size_t)krow * NH + hcol] = aR0[r] * wr - aI0[r] * wi;
            ziB[(size_t)krow * NH + hcol] = aR0[r] * wi + aI0[r] * wr;
        }
        {
            const int krow = t1 * 16 + m;
            const size_t wix = ((size_t)krow * NH + hcol) * 2;
            const float wr = w[wix], wi = w[wix + 1];
            zrB[(size_t)krow * NH + hcol] = aR1[r] * wr - aI1[r] * wi;
            ziB[(size_t)krow * NH + hcol] = aR1[r] * wi + aI1[r] * wr;
        }
    }

    // ---- Nyquist row k=256: Xr = sum_s (-1)^s x[s,h]/sqrt(S), Xi = 0 ----
    const int grp = tid >> 4;            // 16 groups x 16 h columns
    const int hh  = tid & 15;
    float p = 0.0f;
#pragma unroll
    for (int j = 0; j < 32; ++j)
        p += xs[(grp + (j << 4)) * 16 + hh];     // s = grp + 16j, parity = grp&1
    red[grp * 16 + hh] = (grp & 1) ? -p : p;
    __syncthreads();
    if (tid < 16) {
        float tot = 0.0f;
#pragma unroll
        for (int g = 0; g < 16; ++g) tot += red[g * 16 + tid];
        const float xr = tot * ORTHO_SCALE;
        const int h = h0 + tid;
        const size_t wix = ((size_t)256 * NH + h) * 2;
        zrB[(size_t)256 * NH + h] = xr * w[wix];
        ziB[(size_t)256 * NH + h] = xr * w[wix + 1];
    }
    if (tid < 112) {                     // zero pad rows 257..263 (ws is poisoned)
        const int row = 257 + (tid >> 4);
        const int h   = h0 + (tid & 15);
        zrB[(size_t)row * NH + h] = 0.0f;
        ziB[(size_t)row * NH + h] = 0.0f;
    }
}

// ---------------------------------------------------------------------------
// Pipelined half of stage C: acc += G_part @ Z_part for this wave's h pair.
// ---------------------------------------------------------------------------
__device__ __forceinline__ void inv_gemm_part(const float* __restrict__ pga,
                                              const float* __restrict__ zB,
                                              int cbase, int h0a, int h0b,
                                              v8f& acc0, v8f& acc1) {
    v4f A = *(const v4f*)(pga);
    float b0a = zB[cbase + h0a],          b1a = zB[cbase + NH + h0a];
    float b0b = zB[cbase + h0b],          b1b = zB[cbase + NH + h0b];
    float c0a = zB[cbase + 4 * NH + h0a], c1a = zB[cbase + 5 * NH + h0a];
    float c0b = zB[cbase + 4 * NH + h0b], c1b = zB[cbase + 5 * NH + h0b];

    for (int i = 0; i < 33; ++i) {
        const int ip = (i < 32) ? i + 1 : 32;        // uniform clamp
        v4f nA = *(const v4f*)(pga + ip * 128);
        const size_t nb = (size_t)cbase + (size_t)ip * 8 * NH;
        const float nb0a = zB[nb + h0a],          nb1a = zB[nb + NH + h0a];
        const float nb0b = zB[nb + h0b],          nb1b = zB[nb + NH + h0b];
        const float nc0a = zB[nb + 4 * NH + h0a], nc1a = zB[nb + 5 * NH + h0a];
        const float nc0b = zB[nb + 4 * NH + h0b], nc1b = zB[nb + 5 * NH + h0b];

        const v2f a01 = __builtin_shufflevector(A, A, 0, 1);
        const v2f a23 = __builtin_shufflevector(A, A, 2, 3);
        v2f Bf;
        Bf.x = b0a; Bf.y = b1a; acc0 = WMMA_F32(a01, Bf, acc0);
        Bf.x = b0b; Bf.y = b1b; acc1 = WMMA_F32(a01, Bf, acc1);
        Bf.x = c0a; Bf.y = c1a; acc0 = WMMA_F32(a23, Bf, acc0);
        Bf.x = c0b; Bf.y = c1b; acc1 = WMMA_F32(a23, Bf, acc1);

        A = nA;
        b0a = nb0a; b1a = nb1a; b0b = nb0b; b1b = nb1b;
        c0a = nc0a; c1a = nc1a; c0b = nc0b; c1b = nc1b;
    }
}

// ---------------------------------------------------------------------------
// Stage C: y = Gc @ Zr + Gs @ Zi per batch; fused residual + LayerNorm over H.
// ---------------------------------------------------------------------------
__global__ __launch_bounds__(256) void stage_c(const float* __restrict__ x,
                                               const float* __restrict__ gcS,
                                               const float* __restrict__ gsS,
                                               const float* __restrict__ zr,
                                               const float* __restrict__ zi,
                                               const float* __restrict__ gamma,
                                               const float* __restrict__ beta,
                                               float* __restrict__ out,
                                               int b0) {
    __shared__ float smem[16 * NH];      // 16 KB (s,h) staging tile

    const int sT   = blockIdx.x;         // 0..31 s tile
    const int bb   = blockIdx.y;         // 0..CHUNK-1
    const int b    = b0 + bb;
    const int tid  = threadIdx.x;
    const int wv   = tid >> 5;
    const int lane = tid & 31;
    const int lm   = lane & 15;
    const int hi   = lane >> 4;

    const int s0 = sT * 16;
    // prefetch residual rows used in the epilogue (16 KB, one line per thread)
    __builtin_prefetch(x + ((size_t)b * SEQ + s0) * NH + tid * 16, 0, 1);

    const float* __restrict__ pgc = gcS + (size_t)sT * 4224 + (size_t)lane * 4;
    const float* __restrict__ pgs = gsS + (size_t)sT * 4224 + (size_t)lane * 4;
    const float* __restrict__ zrB = zr + (size_t)bb * KR * NH;
    const float* __restrict__ ziB = zi + (size_t)bb * KR * NH;

    const int h0a   = (2 * wv) * 16;
    const int h0b   = h0a + 16;
    const int cbase = 2 * hi * NH + lm;

    v8f acc0 = {0.f,0.f,0.f,0.f,0.f,0.f,0.f,0.f};
    v8f acc1 = {0.f,0.f,0.f,0.f,0.f,0.f,0.f,0.f};

    inv_gemm_part(pgc, zrB, cbase, h0a, h0b, acc0, acc1);   // real part
    inv_gemm_part(pgs, ziB, cbase, h0a, h0b, acc0, acc1);   // imag part

    // stage accumulators into LDS: seq_emb tile [16][256]
#pragma unroll
    for (int r = 0; r < 8; ++r) {
        const int m = r + 8 * hi;
        smem[m * NH + h0a + lm] = acc0[r];
        smem[m * NH + h0b + lm] = acc1[r];
    }
    __syncthreads();

    // residual + LayerNorm over H: wave wv owns rows 2wv and 2wv+1
#pragma unroll
    for (int rr = 0; rr < 2; ++rr) {
        const int row = 2 * wv + rr;
        const int s   = s0 + row;
        const float* __restrict__ xp = x + ((size_t)b * SEQ + s) * NH;
        float vals[8];
        float sum = 0.0f, sq = 0.0f;
#pragma unroll
        for (int j = 0; j < 8; ++j) {
            const int h = lane + 32 * j;
            const float v = smem[row * NH + h] + xp[h];
            vals[j] = v;
            sum += v;
            sq  += v * v;
        }
#pragma unroll
        for (int off = 16; off > 0; off >>= 1) {
            sum += __shfl_xor(sum, off);
            sq  += __shfl_xor(sq, off);
        }
        const float mean = sum * (1.0f / (float)NH);
        const float var  = sq * (1.0f / (float)NH) - mean * mean;
        const float rs   = rsqrtf(var + 1e-12f);
        float* __restrict__ op = out + ((size_t)b * SEQ + s) * NH;
#pragma unroll
        for (int j = 0; j < 8; ++j) {
            const int h = lane + 32 * j;
            op[h] = (vals[j] - mean) * rs * gamma[h] + beta[h];
        }
    }
}

// ---------------------------------------------------------------------------
extern "C" void kernel_launch(void* const* d_in, const int* in_sizes, int n_in,
                              void* d_out, int out_size, void* d_ws, size_t ws_size,
                              hipStream_t stream) {
    const float* x     = (const float*)d_in[0];   // [256,512,256]
    const float* w     = (const float*)d_in[1];   // [1,257,256,2]
    const float* gamma = (const float*)d_in[2];   // [256]
    const float* beta  = (const float*)d_in[3];   // [256]
    float* out = (float*)d_out;

    // ws layout (floats): FcS,FsS [131072]; GcS,GsS [135168];
    // Zr,Zi [CHUNK*264*256]  -> ~19.4 MB total
    float* ws  = (float*)d_ws;
    float* fcS = ws;
    float* fsS = fcS + FWD_ELEMS;
    float* gcS = fsS + FWD_ELEMS;
    float* gsS = gcS + INV_ELEMS;
    float* zr  = gsS + INV_ELEMS;
    float* zi  = zr + (size_t)CHUNK * KR * NH;

    build_dft<<<(INV_ELEMS + 255) / 256, 256, 0, stream>>>(fcS, fsS, gcS, gsS);

    for (int b0 = 0; b0 < NB; b0 += CHUNK) {
        stage_a<<<dim3(NH / 16, CHUNK), 256, 0, stream>>>(x, w, fcS, fsS, zr, zi, b0);
        stage_c<<<dim3(SEQ / 16, CHUNK), 256, 0, stream>>>(x, gcS, gsS, zr, zi,
                                                           gamma, beta, out, b0);
    }
}